// SceneEncoder_33191507263970
// MI455X (gfx1250) — compile-verified
//
#include <hip/hip_runtime.h>
#include <hip/hip_bf16.h>

#define Hdim 128

typedef __attribute__((ext_vector_type(2))) float v2f;
typedef __attribute__((ext_vector_type(8))) float v8f;

// ---------------- input linear: h[n,o] = b[o] + sum_k x[n,k]*W[o,k] ----------------
__global__ void in_linear_kernel(const float* __restrict__ x, const float* __restrict__ W,
                                 const float* __restrict__ b, float* __restrict__ h,
                                 int N, int K) {
    int i = blockIdx.x * blockDim.x + threadIdx.x;
    if (i >= N * Hdim) return;
    int n = i >> 7, o = i & 127;
    float acc = b[o];
    for (int k = 0; k < K; ++k) acc += x[(size_t)n * K + k] * W[(size_t)o * K + k];
    h[i] = acc;
}

// ---------------- edge-parallel gather + atomic scatter-add ----------------
__global__ void scatter_add_kernel(const float* __restrict__ hsrc, const int* __restrict__ ei,
                                   float* __restrict__ agg, float* __restrict__ cnt, int E) {
    int i = blockIdx.x * blockDim.x + threadIdx.x;
    int e = i >> 5;                       // 32 threads per edge (128 feats / float4)
    if (e >= E) return;
    int c   = (i & 31) * 4;
    int src = ei[e];
    int dst = ei[E + e];
    const float4 v = *(const float4*)(hsrc + (size_t)src * Hdim + c);
    float* p = agg + (size_t)dst * Hdim + c;
    atomicAdd(p + 0, v.x);
    atomicAdd(p + 1, v.y);
    atomicAdd(p + 2, v.z);
    atomicAdd(p + 3, v.w);
    if ((i & 31) == 0) atomicAdd(cnt + dst, 1.0f);
}

// ---------------- WMMA GEMM: acc[n,o] += sum_k (A[n,k]*scale[n]) * W[o,k] (+ bias[o]) ----
// One wave owns one 16-row node tile and all 8 output tiles (8 v8f accumulators = 64 VGPRs).
// V_WMMA_F32_16X16X4_F32, K-loop step 4; A fragment loaded once per K-step, reused 8x.
// f32 A layout (16x4): lanes 0-15 hold K={k,k+1}, lanes 16-31 hold K={k+2,k+3} (M = lane%16).
// f32 B layout (4x16): symmetric, N = lane%16. B[k][o] = W[o][k] -> same float2 pattern.
// Optional cnt: the SAGE mean-divide 1/max(cnt[row],1) depends only on the A row (M=lane%16),
// so it is folded into the A fragment as a single multiply (removes a 1.6 GB divide pass).
__global__ void __launch_bounds__(256) mm_acc_kernel(const float* __restrict__ A,
                                                     const float* __restrict__ W,
                                                     const float* __restrict__ bias,
                                                     const float* __restrict__ cnt,
                                                     float* __restrict__ acc, int NP) {
    int wave = (blockIdx.x * blockDim.x + threadIdx.x) >> 5;
    int lane = threadIdx.x & 31;
    int ntiles = NP >> 4;
    if (wave >= ntiles) return;   // wave-uniform: EXEC stays all-ones for WMMA
    int nt = wave;

    int half = lane >> 4;         // 0 or 1 (selects K pair)
    int l16  = lane & 15;         // A row within tile / B column within tile

    float scale = 1.0f;
    if (cnt) {
        float cv = cnt[nt * 16 + l16];
        scale = 1.0f / (cv > 1.0f ? cv : 1.0f);
    }

    const float* aRow  = A + ((size_t)(nt * 16 + l16)) * Hdim + half * 2;
    const float* wBase = W + (size_t)l16 * Hdim + half * 2;   // row o = ot*16 + l16

    v8f c[8] = {};
#pragma unroll 2
    for (int k = 0; k < Hdim; k += 4) {
        v2f a = *(const v2f*)(aRow + k);
        a = a * scale;
#pragma unroll
        for (int ot = 0; ot < 8; ++ot) {
            v2f b = *(const v2f*)(wBase + (size_t)ot * 16 * Hdim + k);
            c[ot] = __builtin_amdgcn_wmma_f32_16x16x4_f32(false, a, false, b, (short)0, c[ot],
                                                          false, false);
        }
    }

    // D layout: VGPR r -> row (half ? r+8 : r), col = l16
    float* outBase = acc + ((size_t)(nt * 16 + half * 8)) * Hdim + l16;
#pragma unroll
    for (int ot = 0; ot < 8; ++ot) {
        float bo = bias ? bias[ot * 16 + l16] : 0.0f;
        float* out = outBase + ot * 16;
#pragma unroll
        for (int r = 0; r < 8; ++r) out[(size_t)r * Hdim] += c[ot][r] + bo;
    }
}

// ---------------- sum of 4 Wr matrices (root-lin merge across edge types) ----------------
__global__ void sum4_kernel(const float* __restrict__ a, const float* __restrict__ b,
                            const float* __restrict__ c, const float* __restrict__ d,
                            float* __restrict__ o, int n) {
    int i = blockIdx.x * blockDim.x + threadIdx.x;
    if (i < n) o[i] = a[i] + b[i] + c[i] + d[i];
}

// ---------------- relu(acc) -> x ----------------
__global__ void relu_copy_kernel(const float* __restrict__ a, float* __restrict__ x, int n) {
    int i = blockIdx.x * blockDim.x + threadIdx.x;
    if (i < n) x[i] = fmaxf(a[i], 0.0f);
}

// ---------------- per-column partial sums over N rows ----------------
__global__ void colsum_kernel(const float* __restrict__ h, float* __restrict__ sum, int N) {
    int col = threadIdx.x;  // 128 threads
    int rowsPerBlock = (N + gridDim.x - 1) / gridDim.x;
    int r0 = blockIdx.x * rowsPerBlock;
    int r1 = r0 + rowsPerBlock; if (r1 > N) r1 = N;
    float acc = 0.0f;
    for (int r = r0; r < r1; ++r) acc += h[(size_t)r * Hdim + col];
    if (r1 > r0) atomicAdd(&sum[col], acc);
}

// ---------------- head: mean combine, layernorm, MLP 128->128->128 ----------------
__global__ void head_kernel(const float* __restrict__ sumW, const float* __restrict__ sumB,
                            float invN, const float* __restrict__ ln_g, const float* __restrict__ ln_b,
                            const float* __restrict__ o1W, const float* __restrict__ o1b,
                            const float* __restrict__ o2W, const float* __restrict__ o2b,
                            float* __restrict__ out) {
    __shared__ float s[Hdim];
    __shared__ float t[Hdim];
    __shared__ float mu, rstd;
    int i = threadIdx.x;
    float v = (sumW[i] * invN + sumB[i] * invN) * 0.5f;
    s[i] = v;
    __syncthreads();
    if (i == 0) {
        float m = 0.0f;
        for (int k = 0; k < Hdim; ++k) m += s[k];
        m /= (float)Hdim;
        float var = 0.0f;
        for (int k = 0; k < Hdim; ++k) { float d = s[k] - m; var += d * d; }
        var /= (float)Hdim;
        mu = m; rstd = rsqrtf(var + 1e-5f);
    }
    __syncthreads();
    float hn = (v - mu) * rstd * ln_g[i] + ln_b[i];
    s[i] = hn;
    __syncthreads();
    float a = o1b[i];
    for (int k = 0; k < Hdim; ++k) a += s[k] * o1W[(size_t)i * Hdim + k];
    a = fmaxf(a, 0.0f);
    t[i] = a;
    __syncthreads();
    float o = o2b[i];
    for (int k = 0; k < Hdim; ++k) o += t[k] * o2W[(size_t)i * Hdim + k];
    out[i] = o;
}

extern "C" void kernel_launch(void* const* d_in, const int* in_sizes, int n_in,
                              void* d_out, int out_size, void* d_ws, size_t ws_size,
                              hipStream_t stream) {
    (void)out_size; (void)ws_size;
    const int N = in_sizes[0] / 5;
    const int E = in_sizes[2] / 2;
    const int NP = (N + 15) & ~15;   // pad node rows to multiple of 16 for WMMA tiles

    const int* ei[8];
    for (int tt = 0; tt < 8; ++tt) ei[tt] = (const int*)d_in[2 + tt];

    // -------- resolve parameter pointers (3 possible flattening schemes) --------
    const float *Wl[2][8], *Wr[2][8], *bl[2][8];
    const float *inWW, *inWb, *inBW, *inBb, *lng, *lnb, *o1W, *o1b, *o2W, *o2b;

    if (n_in >= 68) {
        auto rdType = [&](int base, int l, int tcanon) {
            Wl[l][tcanon] = (const float*)d_in[base];
            if (in_sizes[base + 1] == Hdim) {      // order Wl, bl, Wr
                bl[l][tcanon] = (const float*)d_in[base + 1];
                Wr[l][tcanon] = (const float*)d_in[base + 2];
            } else {                               // order Wl, Wr, bl (sorted keys)
                Wr[l][tcanon] = (const float*)d_in[base + 1];
                bl[l][tcanon] = (const float*)d_in[base + 2];
            }
        };
        if (in_sizes[10] == Hdim * Hdim) {
            // jax-pytree sorted-key order: 'convs' leaves first, edge names sorted = reverse canonical
            int p = 10;
            for (int l = 0; l < 2; ++l)
                for (int srt = 0; srt < 8; ++srt) { rdType(p, l, 7 - srt); p += 3; }
            inBW = (const float*)d_in[p++]; inBb = (const float*)d_in[p++];
            inWW = (const float*)d_in[p++]; inWb = (const float*)d_in[p++];
            lnb  = (const float*)d_in[p++]; lng  = (const float*)d_in[p++];
            o1W  = (const float*)d_in[p++]; o1b  = (const float*)d_in[p++];
            o2W  = (const float*)d_in[p++]; o2b  = (const float*)d_in[p++];
        } else {
            // insertion order: in_wall_W/b, in_ball_W/b, convs (canonical order), ln_g, ln_b, o1, o2
            inWW = (const float*)d_in[10]; inWb = (const float*)d_in[11];
            inBW = (const float*)d_in[12]; inBb = (const float*)d_in[13];
            int p = 14;
            for (int l = 0; l < 2; ++l)
                for (int tc = 0; tc < 8; ++tc) { rdType(p, l, tc); p += 3; }
            lng = (const float*)d_in[p++]; lnb = (const float*)d_in[p++];
            o1W = (const float*)d_in[p++]; o1b = (const float*)d_in[p++];
            o2W = (const float*)d_in[p++]; o2b = (const float*)d_in[p++];
        }
    } else {
        // single concatenated params blob, insertion order
        const float* P = (const float*)d_in[10];
        inWW = P; P += Hdim * 5; inWb = P; P += Hdim;
        inBW = P; P += Hdim * 3; inBb = P; P += Hdim;
        for (int l = 0; l < 2; ++l)
            for (int tc = 0; tc < 8; ++tc) {
                Wl[l][tc] = P; P += Hdim * Hdim;
                bl[l][tc] = P; P += Hdim;
                Wr[l][tc] = P; P += Hdim * Hdim;
            }
        lng = P; P += Hdim; lnb = P; P += Hdim;
        o1W = P; P += Hdim * Hdim; o1b = P; P += Hdim;
        o2W = P; P += Hdim * Hdim; o2b = P; P += Hdim;
    }

    // -------- workspace carve --------
    float* w = (float*)d_ws;
    float* xW = w;        w += (size_t)NP * Hdim;
    float* xB = w;        w += (size_t)NP * Hdim;
    float* accA[2];
    accA[0] = w;          w += (size_t)NP * Hdim;
    accA[1] = w;          w += (size_t)NP * Hdim;
    float* agg = w;       w += (size_t)NP * Hdim;
    float* cnt = w;       w += (size_t)NP;
    float* WrS = w;       w += (size_t)Hdim * Hdim;
    float* sumW = w;      w += Hdim;
    float* sumB = w;      w += Hdim;

    const size_t nElems = (size_t)NP * Hdim;
    const int blocksNE = (int)((nElems + 255) / 256);
    const int mmWaves  = NP / 16;                      // one wave per 16-row node tile
    const int mmBlocks = (mmWaves * 32 + 255) / 256;

    // zero pad tail of feature tables (pad rows must be defined, values never escape)
    if (NP > N) {
        hipMemsetAsync(xW + (size_t)N * Hdim, 0, (size_t)(NP - N) * Hdim * sizeof(float), stream);
        hipMemsetAsync(xB + (size_t)N * Hdim, 0, (size_t)(NP - N) * Hdim * sizeof(float), stream);
    }

    // input encoders
    in_linear_kernel<<<(N * Hdim + 255) / 256, 256, 0, stream>>>((const float*)d_in[0], inWW, inWb, xW, N, 5);
    in_linear_kernel<<<(N * Hdim + 255) / 256, 256, 0, stream>>>((const float*)d_in[1], inBW, inBb, xB, N, 3);

    float* x[2] = {xW, xB};
    for (int l = 0; l < 2; ++l) {
        hipMemsetAsync(accA[0], 0, nElems * sizeof(float), stream);
        hipMemsetAsync(accA[1], 0, nElems * sizeof(float), stream);
        for (int tc = 0; tc < 8; ++tc) {
            int s = tc >> 2, d = tc & 1;   // canonical: t = s*4 + r*2 + d
            hipMemsetAsync(agg, 0, nElems * sizeof(float), stream);
            hipMemsetAsync(cnt, 0, (size_t)NP * sizeof(float), stream);
            scatter_add_kernel<<<(E * 32 + 255) / 256, 256, 0, stream>>>(x[s], ei[tc], agg, cnt, E);
            // GEMM with fused mean-divide (cnt) and bias
            mm_acc_kernel<<<mmBlocks, 256, 0, stream>>>(agg, Wl[l][tc], bl[l][tc], cnt, accA[d], NP);
        }
        // merged root linear: x_d @ (sum of the 4 Wr with dst=d)^T
        for (int d = 0; d < 2; ++d) {
            sum4_kernel<<<(Hdim * Hdim + 255) / 256, 256, 0, stream>>>(
                Wr[l][d], Wr[l][2 + d], Wr[l][4 + d], Wr[l][6 + d], WrS, Hdim * Hdim);
            mm_acc_kernel<<<mmBlocks, 256, 0, stream>>>(x[d], WrS, nullptr, nullptr, accA[d], NP);
        }
        relu_copy_kernel<<<blocksNE, 256, 0, stream>>>(accA[0], xW, (int)nElems);
        relu_copy_kernel<<<blocksNE, 256, 0, stream>>>(accA[1], xB, (int)nElems);
    }

    hipMemsetAsync(sumW, 0, Hdim * sizeof(float), stream);
    hipMemsetAsync(sumB, 0, Hdim * sizeof(float), stream);
    colsum_kernel<<<512, Hdim, 0, stream>>>(xW, sumW, N);
    colsum_kernel<<<512, Hdim, 0, stream>>>(xB, sumB, N);
    head_kernel<<<1, Hdim, 0, stream>>>(sumW, sumB, 1.0f / (float)N, lng, lnb,
                                        o1W, o1b, o2W, o2b, (float*)d_out);
}